// MulticlassBoundaryLoss_89970974916940
// MI455X (gfx1250) — compile-verified
//
#include <hip/hip_runtime.h>
#include <hip/hip_bf16.h>
#include <stdint.h>

// Problem constants (from reference)
#define Bc   8
#define Cc   19
#define Hc   256
#define Wc   512
#define Rr   20
#define TH   16
#define TW   32
#define HWH  (TH + 2*Rr)   // 56 halo rows
#define WWH  (TW + 2*Rr)   // 72 halo cols
#define NPIX (TH*TW)       // 512
#define NTH  256

// Dynamic LDS layout (bytes)
#define OFF_LOG   0
#define SZ_LOG    (Cc*TH*TW*4)            // 38912: logits tile [c][h][w]
#define OFF_GNEG  (OFF_LOG + SZ_LOG)      // 38912
#define SZ_GNEG   (Cc*TH*WWH*2)           // 43776: pass1 background dist^2 (u16)
#define OFF_MAX   (OFF_GNEG + SZ_GNEG)    // 82688
#define SZ_MAX    (NPIX*4)
#define OFF_DINV  (OFF_MAX + SZ_MAX)      // 84736
#define SZ_DINV   (NPIX*4)
#define OFF_RED   (OFF_DINV + SZ_DINV)    // 86784
#define SZ_RED    (32)
#define OFF_T     (OFF_RED + SZ_RED)      // 86816
#define SZ_T      (HWH*WWH)               // 4032: target classes u8 (255 = OOB)
#define OFF_DD    (OFF_T + SZ_T)          // 90848
#define SZ_DD     (TH*WWH)                // 1152: nearest-differing dist u8 (255 = none)
#define LDS_BYTES (OFF_DD + SZ_DD)        // 92000

typedef unsigned int u32;
typedef __attribute__((ext_vector_type(4))) u32 v4u;
typedef __attribute__((ext_vector_type(8))) u32 v8u;

__device__ __forceinline__ int imin(int a, int b) { return a < b ? a : b; }

__global__ void __launch_bounds__(NTH)
boundary_loss_kernel(const float* __restrict__ logits,
                     const int* __restrict__ targets,
                     float* __restrict__ out)
{
    extern __shared__ char smem[];
    float*          shLog  = (float*)(smem + OFF_LOG);
    unsigned short* shGneg = (unsigned short*)(smem + OFF_GNEG);
    float*          shMax  = (float*)(smem + OFF_MAX);
    float*          shDinv = (float*)(smem + OFF_DINV);
    float*          shRed  = (float*)(smem + OFF_RED);
    unsigned char*  shT    = (unsigned char*)(smem + OFF_T);
    unsigned char*  shDd   = (unsigned char*)(smem + OFF_DD);

    const int tid = threadIdx.x;
    const int w0  = blockIdx.x * TW;
    const int h0  = blockIdx.y * TH;
    const int b   = blockIdx.z;

    // ------------------------------------------------------------------
    // Phase 0: TDM async load of logits tile [C=19][TH=16][TW=32] -> LDS
    // 3-D tensor descriptor: x=TW elems, y=TH rows (stride W), z=C planes
    // (stride H*W). data_size = 4 bytes. Issued once by wave 0; TDM
    // ignores EXEC and is tracked with TENSORcnt.
    // ------------------------------------------------------------------
    {
        unsigned long long ga = (unsigned long long)(uintptr_t)
            (logits + ((size_t)b * Cc * Hc * Wc + (size_t)h0 * Wc + (size_t)w0));
        u32 ldsa = (u32)(uintptr_t)shLog;   // low 32 bits of flat addr = LDS byte offset

        v4u g0;
        g0[0] = 1u;                                   // count=1, user descriptor
        g0[1] = ldsa;                                 // lds_addr
        g0[2] = (u32)ga;                              // global_addr[31:0]
        g0[3] = ((u32)(ga >> 32) & 0x1FFFFFFu)        // global_addr[56:32]
              | (2u << 30);                           // type=2 ("image")

        v8u g1;
        g1[0] = (2u << 16);                           // wg_mask=0, data_size=2 (4B)
        g1[1] = ((u32)(Wc & 0xFFFF)) << 16;           // tensor_dim0[15:0]=512
        g1[2] = (((u32)(Hc & 0xFFFF)) << 16)          // tensor_dim1[15:0]=256
              | ((u32)(Wc >> 16) & 0xFFFFu);          // tensor_dim0[31:16]=0
        g1[3] = (((u32)TW) << 16)                     // tile_dim0=32
              | ((u32)(Hc >> 16) & 0xFFFFu);          // tensor_dim1[31:16]=0
        g1[4] = (((u32)Cc) << 16) | ((u32)TH);        // tile_dim2=19, tile_dim1=16
        g1[5] = (u32)Wc;                              // tensor_dim0_stride[31:0]=512
        g1[6] = (((u32)((Hc * Wc) & 0xFFFF)) << 16);  // dim1_stride[15:0]=0 | dim0_stride[47:32]=0
        g1[7] = (u32)((Hc * Wc) >> 16);               // dim1_stride[47:16]=2 (131072)

        v4u g2; g2[0] = (u32)Cc; g2[1] = 0; g2[2] = 0; g2[3] = 0;  // tensor_dim2=19
        v4u g3; g3[0] = 0; g3[1] = 0; g3[2] = 0; g3[3] = 0;

        if (tid < 32) {
            asm volatile("tensor_load_to_lds %0, %1, %2, %3"
                         :: "s"(g0), "s"(g1), "s"(g2), "s"(g3)
                         : "memory");
        }
    }

    // ------------------------------------------------------------------
    // Phase 1 (overlapped with TDM): targets halo -> LDS u8; init gneg=INF
    // ------------------------------------------------------------------
    for (int i = tid; i < HWH * WWH; i += NTH) {
        int hh = i / WWH, xx = i - hh * WWH;
        int gh = h0 - Rr + hh;
        int gw = w0 - Rr + xx;
        unsigned char v = 255;  // OOB sentinel (== INF in the EDT)
        if (gh >= 0 && gh < Hc && gw >= 0 && gw < Wc)
            v = (unsigned char)targets[((size_t)b * Hc + gh) * Wc + gw];
        shT[i] = v;
    }
    {
        u32* g32 = (u32*)shGneg;
        for (int i = tid; i < (Cc * TH * WWH) / 2; i += NTH) g32[i] = 0xFFFFFFFFu;
    }
    __syncthreads();

    // ------------------------------------------------------------------
    // Phase 2: pass-1 along H (window +-R) for rows [0,TH) x halo cols.
    //   foreground: dd = nearest in-bounds differing label (class-indep)
    //   background: per-class first-hit distance via 19-bit seen mask
    // ------------------------------------------------------------------
    for (int i = tid; i < TH * WWH; i += NTH) {
        int h = i / WWH, x = i - h * WWH;
        const unsigned char* col = shT + x;
        unsigned tc = col[(h + Rr) * WWH];
        unsigned seen = 0;
        int ddv = 255;
        if (tc < Cc) { seen = 1u << tc; shGneg[(tc * TH + h) * WWH + x] = 0; }
        for (int d = 1; d <= Rr; ++d) {
            unsigned a = col[(h + Rr - d) * WWH];
            unsigned e = col[(h + Rr + d) * WWH];
            unsigned short d2 = (unsigned short)(d * d);
            if (a < Cc && !((seen >> a) & 1u)) { seen |= 1u << a; shGneg[(a * TH + h) * WWH + x] = d2; }
            if (e < Cc && !((seen >> e) & 1u)) { seen |= 1u << e; shGneg[(e * TH + h) * WWH + x] = d2; }
            if (ddv == 255 && ((a < Cc && a != tc) || (e < Cc && e != tc))) ddv = d;
        }
        shDd[i] = (unsigned char)ddv;
    }

    __builtin_amdgcn_s_wait_tensorcnt(0);   // logits tile resident in LDS
    __syncthreads();

    // ------------------------------------------------------------------
    // Phase 3: per-pixel softmax stats from LDS logits (read global once)
    // ------------------------------------------------------------------
    for (int p = tid; p < NPIX; p += NTH) {
        float m = -3.0e38f;
        for (int c = 0; c < Cc; ++c) m = fmaxf(m, shLog[c * NPIX + p]);
        float s = 0.f;
        for (int c = 0; c < Cc; ++c) s += __expf(shLog[c * NPIX + p] - m);
        shMax[p]  = m;
        shDinv[p] = 1.0f / s;
    }
    __syncthreads();

    // ------------------------------------------------------------------
    // Phase 4: pass-2 along W + sdf + probs*sdf accumulation
    // ------------------------------------------------------------------
    float acc = 0.f;
    for (int p = tid; p < NPIX; p += NTH) {
        int h = p / TW, w = p - (p / TW) * TW;
        const unsigned char* trow = shT  + (h + Rr) * WWH + w;  // labels along window
        const unsigned char* drow = shDd +  h       * WWH + w;  // dd along window

        // Hoist window data into packed registers (class-independent).
        u32 tp[11];
        #pragma unroll
        for (int q = 0; q < 11; ++q) {
            u32 v = 0;
            #pragma unroll
            for (int j = 0; j < 4; ++j) {
                int e = q * 4 + j;
                u32 byte = (e < 2 * Rr + 1) ? (u32)trow[e] : 255u;
                v |= byte << (8 * j);
            }
            tp[q] = v;
        }
        u32 pvp[21];  // posval[e] = (dd==255) ? BIG : dd^2  (u16 pairs)
        #pragma unroll
        for (int q = 0; q < 21; ++q) {
            u32 v = 0;
            #pragma unroll
            for (int j = 0; j < 2; ++j) {
                int e = q * 2 + j;
                u32 pd = 1023u;
                if (e < 2 * Rr + 1) { u32 dd = drow[e]; pd = (dd == 255u) ? 1023u : dd * dd; }
                v |= pd << (16 * j);
            }
            pvp[q] = v;
        }

        float m = shMax[p], dinv = shDinv[p];
        for (int c = 0; c < Cc; ++c) {
            int base = (c * TH + h) * WWH + w;
            int mpos = 1 << 20, mneg = 1 << 20;
            #pragma unroll
            for (int e = 0; e <= 2 * Rr; ++e) {
                const int d2 = (e - Rr) * (e - Rr);
                u32 t  = (tp[e >> 2] >> ((e & 3) * 8)) & 255u;
                u32 pv = (pvp[e >> 1] >> ((e & 1) * 16)) & 0xFFFFu;
                int gpos = (t == (u32)c) ? (int)pv : ((t == 255u) ? 1023 : 0);
                mpos = imin(mpos, d2 + gpos);
                int gn = (int)shGneg[base + e];      // ds_load_u16, imm offset
                mneg = imin(mneg, d2 + gn);
            }
            float pd  = sqrtf((float)imin(mpos, Rr * Rr));
            float nd  = sqrtf((float)imin(mneg, Rr * Rr));
            float sdf = (nd - pd) * (1.0f / (float)Rr);   // clip(-R,R)/R is implicit
            float prob = __expf(shLog[c * NPIX + p] - m) * dinv;
            acc += prob * sdf;
        }
    }

    // ------------------------------------------------------------------
    // Phase 5: reduce (wave32 shuffle -> LDS -> atomic)
    // ------------------------------------------------------------------
    for (int off = 16; off > 0; off >>= 1)
        acc += __shfl_down(acc, off, 32);
    int lane = tid & 31, wv = tid >> 5;
    if (lane == 0) shRed[wv] = acc;
    __syncthreads();
    if (tid == 0) {
        float s = 0.f;
        for (int i = 0; i < NTH / 32; ++i) s += shRed[i];
        atomicAdd(out, s * (1.0f / ((float)Bc * (float)Hc * (float)Wc)));
    }
}

__global__ void zero_out_kernel(float* out) { out[0] = 0.0f; }

extern "C" void kernel_launch(void* const* d_in, const int* in_sizes, int n_in,
                              void* d_out, int out_size, void* d_ws, size_t ws_size,
                              hipStream_t stream) {
    (void)in_sizes; (void)n_in; (void)out_size; (void)d_ws; (void)ws_size;
    const float* logits  = (const float*)d_in[0];
    const int*   targets = (const int*)d_in[1];
    float*       out     = (float*)d_out;

    zero_out_kernel<<<1, 1, 0, stream>>>(out);

    dim3 grid(Wc / TW, Hc / TH, Bc);   // 16 x 16 x 8 = 2048 tiles
    boundary_loss_kernel<<<grid, NTH, LDS_BYTES, stream>>>(logits, targets, out);
}